// EnergyAE_27127013442018
// MI455X (gfx1250) — compile-verified
//
#include <hip/hip_runtime.h>
#include <hip/hip_bf16.h>
#include <math.h>

// ---------------- CDNA5 WMMA (fp32, 16x16x4) ----------------
// D = A(16x4) * B(4x16) + C(16x16), wave32, f32 everywhere.
//  A: lane l (m=l&15): VGPR0=K(kbase+klo), VGPR1=K(kbase+klo+1), klo=(l>=16)?2:0
//  B: lane l (n=l&15): VGPR0=row(kbase+klo), VGPR1=row(kbase+klo+1)
//  C/D: VGPR r: lanes 0-15 -> M=r, lanes 16-31 -> M=r+8; N = lane&15
typedef __attribute__((ext_vector_type(2))) float v2f;
typedef __attribute__((ext_vector_type(8))) float v8f;

__device__ __forceinline__ v8f wmma_f32(v2f a, v2f b, v8f c) {
  return __builtin_amdgcn_wmma_f32_16x16x4_f32(
      /*neg_a=*/false, a, /*neg_b=*/false, b,
      /*c_mod=*/(short)0, c, /*reuse_a=*/false, /*reuse_b=*/false);
}

// ---- CDNA5 async global->LDS DMA (ASYNCcnt-tracked), GV addressing ----
// Generic pointers to __shared__ carry the workgroup-relative LDS offset in
// addr[31:0] (flat-LDS aperture), which is exactly what VDST wants.
__device__ __forceinline__ unsigned lds_off(const void* p) {
  return (unsigned)(unsigned long long)(uintptr_t)p;
}
__device__ __forceinline__ void async_ld_b128(unsigned lds, const void* g) {
  asm volatile("global_load_async_to_lds_b128 %0, %1, off"
               :
               : "v"(lds), "v"((unsigned long long)(uintptr_t)g)
               : "memory");
}
__device__ __forceinline__ void async_ld_b64(unsigned lds, const void* g) {
  asm volatile("global_load_async_to_lds_b64 %0, %1, off"
               :
               : "v"(lds), "v"((unsigned long long)(uintptr_t)g)
               : "memory");
}
__device__ __forceinline__ void wait_async_le(int n) {
  if (n == 0)
    asm volatile("s_wait_asynccnt 0x0" ::: "memory");
  else if (n == 1)
    asm volatile("s_wait_asynccnt 0x1" ::: "memory");
  else
    asm volatile("s_wait_asynccnt 0x2" ::: "memory");
}

#define DD 3072
#define HH 2048
#define NN 16
#define BSZ 512

// ---------------- Generic tiled WMMA GEMM (async double-buffered) ----------
// Block = 256 threads = 8 waves arranged 4(M) x 2(N); WG tile 64x32.
// EPI: 0 = out = tanh(acc+bias)
//      1 = out = aux0 - (acc+bias)              (residual r = x - dec)
//      2 = out = 2*a*(1-a^2)*acc / sigma^2      (c-vector; aux0=a, aux1=sigma)
//      3 = row partial sums of (acc+bias-aux0)^2 -> out[m][n0/16]
template <int EPI, bool TRANSB>
__global__ __launch_bounds__(256) void gemm16(
    const float* __restrict__ A, int lda, const float* __restrict__ B, int ldb,
    const float* __restrict__ bias, float* __restrict__ out,
    const float* __restrict__ aux0, const float* __restrict__ aux1,
    int M, int Ncols, int K) {
  // pitches: 20 floats (80B, 16B-aligned rows, conflict-free for 16-lane m
  // stride) and 34 floats (136B, 8B-aligned rows).
  __shared__ float Al[2][64][20];
  __shared__ float Bl[2][16][34];
  const int tid = threadIdx.x;
  const int lane = tid & 31, wid = tid >> 5;
  const int wm = wid & 3, wn = wid >> 2;
  const int mwg = blockIdx.y * 64;
  const int nwg = blockIdx.x * 32;
  const int lm = lane & 15;
  const int klo = (lane >> 4) << 1;
  // staging index maps
  const int ai = tid >> 2;          // A row 0..63
  const int aj = (tid & 3) << 2;    // A col 0,4,8,12 (b128 each)
  const int bi = tid >> 4;          // B k-row 0..15
  const int bj = (tid & 15) << 1;   // B n-col step 2 (b64 each)
  const int ti = tid >> 3;          // TRANSB: n-local 0..31
  const int tj = (tid & 7) << 1;    // TRANSB: k-local 0..14
  const int NPEND = TRANSB ? 1 : 2; // async ops per wave per panel
  v8f acc = {};

  const int np = K / 16;
  // prologue: panel 0 -> buf 0
  async_ld_b128(lds_off(&Al[0][ai][aj]), A + (mwg + ai) * lda + aj);
  if (!TRANSB) {
    async_ld_b64(lds_off(&Bl[0][bi][bj]), B + bi * ldb + (nwg + bj));
  } else {
    const float* src = B + (nwg + ti) * ldb + tj;
    Bl[0][tj][ti] = src[0];
    Bl[0][tj + 1][ti] = src[1];
  }

  for (int p = 0; p < np; ++p) {
    const int buf = p & 1;
    if (p + 1 < np) {  // issue next panel into the other buffer
      const int k0n = (p + 1) * 16;
      async_ld_b128(lds_off(&Al[buf ^ 1][ai][aj]),
                    A + (mwg + ai) * lda + k0n + aj);
      if (!TRANSB) {
        async_ld_b64(lds_off(&Bl[buf ^ 1][bi][bj]),
                     B + (k0n + bi) * ldb + (nwg + bj));
      } else {
        const float* src = B + (nwg + ti) * ldb + k0n + tj;
        Bl[buf ^ 1][tj][ti] = src[0];
        Bl[buf ^ 1][tj + 1][ti] = src[1];
      }
      wait_async_le(NPEND);  // panel p landed; panel p+1 still in flight
    } else {
      wait_async_le(0);
    }
    __syncthreads();
#pragma unroll
    for (int kk = 0; kk < 16; kk += 4) {
      v2f a, b;
      a.x = Al[buf][wm * 16 + lm][kk + klo];
      a.y = Al[buf][wm * 16 + lm][kk + klo + 1];
      b.x = Bl[buf][kk + klo][wn * 16 + lm];
      b.y = Bl[buf][kk + klo + 1][wn * 16 + lm];
      acc = wmma_f32(a, b, acc);
    }
    __syncthreads();  // all waves done reading buf before it is re-filled
  }

  const int n = nwg + wn * 16 + lm;
  const int mbase = mwg + wm * 16 + ((lane >> 4) << 3);
  if (EPI == 0) {
#pragma unroll
    for (int r = 0; r < 8; ++r)
      out[(mbase + r) * Ncols + n] = tanhf(acc[r] + bias[n]);
  } else if (EPI == 1) {
#pragma unroll
    for (int r = 0; r < 8; ++r) {
      int m = mbase + r;
      out[m * Ncols + n] = aux0[m * Ncols + n] - (acc[r] + bias[n]);
    }
  } else if (EPI == 2) {
#pragma unroll
    for (int r = 0; r < 8; ++r) {
      int m = mbase + r;
      float av = aux0[m * Ncols + n];
      float sg = aux1[m];
      out[m * Ncols + n] = 2.f * av * (1.f - av * av) * acc[r] / (sg * sg);
    }
  } else {  // EPI == 3: fused recon-loss partials
    float sq[8];
#pragma unroll
    for (int r = 0; r < 8; ++r) {
      int m = mbase + r;
      float d = acc[r] + bias[n] - aux0[m * Ncols + n];
      sq[r] = d * d;
    }
#pragma unroll
    for (int r = 0; r < 8; ++r) {  // reduce over 16 n-lanes within each half
      float v = sq[r];
      v += __shfl_xor(v, 1); v += __shfl_xor(v, 2);
      v += __shfl_xor(v, 4); v += __shfl_xor(v, 8);
      sq[r] = v;
    }
    if (lm == 0) {
      int nb = (nwg + wn * 16) >> 4;
      int NB = Ncols >> 4;
#pragma unroll
      for (int r = 0; r < 8; ++r) out[(mbase + r) * NB + nb] = sq[r];
    }
  }
}

// -------- encoder heads: z_star = h@Wmu + bmu ; sigma = exp(h@Wls + bls) ----
__global__ __launch_bounds__(256) void enc_heads(
    const float* __restrict__ h, const float* __restrict__ Wmu,
    const float* __restrict__ bmu, const float* __restrict__ Wls,
    const float* __restrict__ bls, float* __restrict__ z_star,
    float* __restrict__ sigma) {
  __shared__ float Al[128][17];
  __shared__ float sred[256];
  const int tid = threadIdx.x, lane = tid & 31, wid = tid >> 5;
  const int m0 = blockIdx.x * 128;
  const int lm = lane & 15, klo = (lane >> 4) << 1;
  v8f acc = {};
  for (int k0 = 0; k0 < HH; k0 += 16) {
    int i = tid >> 1;
    int j = (tid & 1) << 3;
    const float* src = h + (m0 + i) * HH + k0 + j;
#pragma unroll
    for (int q = 0; q < 8; ++q) Al[i][j + q] = src[q];
    __syncthreads();
#pragma unroll
    for (int kk = 0; kk < 16; kk += 4) {
      v2f a, b;
      a.x = Al[wid * 16 + lm][kk + klo];
      a.y = Al[wid * 16 + lm][kk + klo + 1];
      int krow = k0 + kk + klo;
      b.x = Wmu[krow * NN + lm];
      b.y = Wmu[(krow + 1) * NN + lm];
      acc = wmma_f32(a, b, acc);
    }
    __syncthreads();
  }
  const int mb = m0 + wid * 16 + ((lane >> 4) << 3);
#pragma unroll
  for (int r = 0; r < 8; ++r) z_star[(mb + r) * NN + lm] = acc[r] + bmu[lm];

  // sigma: 2 threads per row
  int row = tid >> 1, half = tid & 1;
  float s = 0.f;
  const float* hr = h + (m0 + row) * HH + half * 1024;
  const float* wl = Wls + half * 1024;
  for (int k = 0; k < 1024; ++k) s += hr[k] * wl[k];
  sred[tid] = s;
  __syncthreads();
  if (!half) sigma[m0 + row] = expf(sred[tid] + sred[tid + 1] + bls[0]);
}

// -------- per-sample Prec = J^T J / sigma^2 + I + W1 diag(c) W1^T ----------
// One workgroup (8 waves) per sample. Two-stage WMMA:
//  stage1: JT tile (16 x 64 per wave) = (W1 .* u) @ W2[:, dcols], K=2048
//  stage2: P1 += JT @ JT^T  (A-operand and B-operand per-lane loads coincide)
__global__ __launch_bounds__(256) void prec_kernel(
    const float* __restrict__ a, const float* __restrict__ cvec,
    const float* __restrict__ W1, const float* __restrict__ W2,
    const float* __restrict__ sigma, float* __restrict__ Prec) {
  __shared__ float u[HH];          // (1 - a^2), reused as reduction buffer
  __shared__ float cl[HH];
  __shared__ float Ap[16][36];     // staged (W1 .* u) K-panel
  __shared__ float JT[8][16][68];  // per-wave JT tile, padded
  const int b = blockIdx.x;
  const int tid = threadIdx.x, lane = tid & 31, wid = tid >> 5;
  const int lm = lane & 15, klo = (lane >> 4) << 1;

  for (int k = tid; k < HH; k += 256) {
    float av = a[b * HH + k];
    u[k] = 1.f - av * av;
    cl[k] = cvec[b * HH + k];
  }
  __syncthreads();

  v8f p1 = {};
  for (int pass = 0; pass < 6; ++pass) {  // D = 6 * 512, wave owns 64 cols
    const int dbase = pass * 512 + wid * 64;
    v8f jt[4] = {{}, {}, {}, {}};
    for (int k0 = 0; k0 < HH; k0 += 32) {
      {  // stage (W1 .* u) panel 16x32
        int i = tid >> 4;
        int j = (tid & 15) << 1;
        float w0 = W1[i * HH + k0 + j], w1 = W1[i * HH + k0 + j + 1];
        Ap[i][j] = w0 * u[k0 + j];
        Ap[i][j + 1] = w1 * u[k0 + j + 1];
      }
      __syncthreads();
#pragma unroll
      for (int kk = 0; kk < 32; kk += 4) {
        v2f av;
        av.x = Ap[lm][kk + klo];
        av.y = Ap[lm][kk + klo + 1];
        int krow = k0 + kk + klo;
        const float* B0 = W2 + krow * DD + dbase + lm;
        const float* B1 = W2 + (krow + 1) * DD + dbase + lm;
#pragma unroll
        for (int s = 0; s < 4; ++s) {
          v2f bv;
          bv.x = B0[s * 16];
          bv.y = B1[s * 16];
          jt[s] = wmma_f32(av, bv, jt[s]);
        }
      }
      __syncthreads();
    }
    // park JT tile in this wave's LDS region (same-wave dependency only)
#pragma unroll
    for (int r = 0; r < 8; ++r) {
      int i = r + ((lane >> 4) << 3);
#pragma unroll
      for (int s = 0; s < 4; ++s) JT[wid][i][s * 16 + lm] = jt[s][r];
    }
    // stage2: P1 += JT @ JT^T ; A and B per-lane loads are identical
#pragma unroll
    for (int kk = 0; kk < 64; kk += 4) {
      v2f v;
      v.x = JT[wid][lm][kk + klo];
      v.y = JT[wid][lm][kk + klo + 1];
      p1 = wmma_f32(v, v, p1);
    }
  }

  // hess partial: (W1 .* c) @ W1^T over this wave's K slice (256 wide)
  v8f hacc = {};
  const int kw0 = wid * 256;
#pragma unroll 4
  for (int kk = 0; kk < 256; kk += 4) {
    int k = kw0 + kk + klo;
    v2f w, av;
    w.x = W1[lm * HH + k];
    w.y = W1[lm * HH + k + 1];
    av.x = w.x * cl[k];
    av.y = w.y * cl[k + 1];
    hacc = wmma_f32(av, w, hacc);
  }

  const float sg = sigma[b];
  const float is2 = 1.f / (sg * sg);
  __syncthreads();           // everyone done with u -> reuse as reduction buf
  float* red = u;            // 8 waves x 256 slots
#pragma unroll
  for (int r = 0; r < 8; ++r)
    red[wid * 256 + r * 32 + lane] = p1[r] * is2 + hacc[r];
  __syncthreads();
  {
    int e = tid;
    float s = 0.f;
#pragma unroll
    for (int w = 0; w < 8; ++w) s += red[w * 256 + e];
    int r = e >> 5, l = e & 31;
    int i = r + ((l & 16) ? 8 : 0), j = l & 15;
    Prec[b * 256 + i * 16 + j] = s + ((i == j) ? 1.f : 0.f);
  }
}

// -------- per-sample eigvals (Jacobi) + Cholesky + triangular solve --------
__global__ __launch_bounds__(32) void eig_chol_kernel(
    const float* __restrict__ Prec, const float* __restrict__ z_star,
    const float* __restrict__ eps, float* __restrict__ z_sample,
    float* __restrict__ lat_e, float* __restrict__ logdet) {
  __shared__ float sm[32 * 257];
  const int t = threadIdx.x;
  const int b = blockIdx.x * 32 + t;
  float* A = &sm[t * 257];
  for (int i = 0; i < 256; ++i) A[i] = Prec[b * 256 + i];
  // cyclic Jacobi, eigenvalues only
  for (int sweep = 0; sweep < 12; ++sweep)
    for (int p = 0; p < 15; ++p)
      for (int q = p + 1; q < 16; ++q) {
        float apq = A[p * 16 + q];
        if (fabsf(apq) < 1e-12f) continue;
        float app = A[p * 16 + p], aqq = A[q * 16 + q];
        float tau = (aqq - app) / (2.f * apq);
        float tt =
            (tau >= 0.f ? 1.f : -1.f) / (fabsf(tau) + sqrtf(1.f + tau * tau));
        float cc = 1.f / sqrtf(1.f + tt * tt);
        float ss = tt * cc;
        for (int k = 0; k < 16; ++k) {
          float x1 = A[p * 16 + k], x2 = A[q * 16 + k];
          A[p * 16 + k] = cc * x1 - ss * x2;
          A[q * 16 + k] = ss * x1 + cc * x2;
        }
        for (int k = 0; k < 16; ++k) {
          float x1 = A[k * 16 + p], x2 = A[k * 16 + q];
          A[k * 16 + p] = cc * x1 - ss * x2;
          A[k * 16 + q] = ss * x1 + cc * x2;
        }
      }
  float emin = A[0];
  for (int i = 1; i < 16; ++i) emin = fminf(emin, A[i * 17]);
  const float delta = 10.f - emin;  // 1/MAX_LATENT_VAR - eigmin
  float inv_s = 0.f, ld = 0.f;
  for (int i = 0; i < 16; ++i) {
    float lam = A[i * 17] + delta;
    inv_s += 0.5f / lam;
    ld += 0.5f * logf(lam);
  }
  // Cholesky of Prec + delta*I (lower, in place)
  for (int i = 0; i < 256; ++i) A[i] = Prec[b * 256 + i];
  for (int i = 0; i < 16; ++i) A[i * 17] += delta;
  for (int i = 0; i < 16; ++i)
    for (int j = 0; j <= i; ++j) {
      float s = A[i * 16 + j];
      for (int k = 0; k < j; ++k) s -= A[i * 16 + k] * A[j * 16 + k];
      A[i * 16 + j] = (i == j) ? sqrtf(s) : s / A[j * 16 + j];
    }
  // solve U y = eps with U = L^T (back substitution)
  float y[16];
  for (int i = 15; i >= 0; --i) {
    float s = eps[b * 16 + i];
    for (int k = i + 1; k < 16; ++k) s -= A[k * 16 + i] * y[k];
    y[i] = s / A[i * 17];
  }
  float ze = 0.f;
  for (int i = 0; i < 16; ++i) {
    float zs = z_star[b * 16 + i];
    z_sample[b * 16 + i] = zs + y[i];
    ze += zs * zs;
  }
  lat_e[b] = 0.5f * ze + inv_s;
  logdet[b] = ld;
}

// -------- finalize: deterministic partial reduce + loss assembly -----------
__global__ __launch_bounds__(256) void finalize_kernel(
    const float* __restrict__ partials, const float* __restrict__ sigma,
    const float* __restrict__ lat_e, const float* __restrict__ logdet,
    float* __restrict__ out) {
  int b = blockIdx.x * 256 + threadIdx.x;
  if (b >= BSZ) return;
  float rs = 0.f;
  for (int nb = 0; nb < (DD / 16); ++nb) rs += partials[b * (DD / 16) + nb];
  float s = sigma[b];
  float recon = rs / (2.f * s * s);
  float lat = lat_e[b], ld = logdet[b];
  float sl = (float)DD * logf(s);
  out[b] = (recon + lat + ld + sl) / (float)DD;
  out[BSZ + b] = recon;
  out[2 * BSZ + b] = lat;
  out[3 * BSZ + b] = ld;
  out[4 * BSZ + b] = s;
}

// --------------------------- launch --------------------------------------
extern "C" void kernel_launch(void* const* d_in, const int* in_sizes, int n_in,
                              void* d_out, int out_size, void* d_ws,
                              size_t ws_size, hipStream_t stream) {
  const float* x = (const float*)d_in[0];
  const float* eps = (const float*)d_in[1];
  const float* enc_W1 = (const float*)d_in[2];
  const float* enc_b1 = (const float*)d_in[3];
  const float* enc_Wmu = (const float*)d_in[4];
  const float* enc_bmu = (const float*)d_in[5];
  const float* enc_Wls = (const float*)d_in[6];
  const float* enc_bls = (const float*)d_in[7];
  const float* dec_W1 = (const float*)d_in[8];
  const float* dec_b1 = (const float*)d_in[9];
  const float* dec_W2 = (const float*)d_in[10];
  const float* dec_b2 = (const float*)d_in[11];
  float* out = (float*)d_out;

  float* ws = (float*)d_ws;
  float* h = ws;        ws += BSZ * HH;
  float* amat = ws;     ws += BSZ * HH;
  float* rres = ws;     ws += BSZ * DD;
  float* cvec = ws;     ws += BSZ * HH;
  float* a2 = ws;       ws += BSZ * HH;
  float* partials = ws; ws += BSZ * (DD / 16);
  float* z_star = ws;   ws += BSZ * NN;
  float* sigma = ws;    ws += BSZ;
  float* Prec = ws;     ws += BSZ * NN * NN;
  float* z_samp = ws;   ws += BSZ * NN;
  float* lat_e = ws;    ws += BSZ;
  float* logd = ws;     ws += BSZ;

  // 1) h = tanh(x @ enc_W1 + b1)
  gemm16<0, false><<<dim3(HH / 32, BSZ / 64), 256, 0, stream>>>(
      x, DD, enc_W1, HH, enc_b1, h, nullptr, nullptr, BSZ, HH, DD);
  // 2) z_star, sigma
  enc_heads<<<dim3(BSZ / 128), 256, 0, stream>>>(h, enc_Wmu, enc_bmu, enc_Wls,
                                                 enc_bls, z_star, sigma);
  // 3) a = tanh(z_star @ dec_W1 + b1)
  gemm16<0, false><<<dim3(HH / 32, BSZ / 64), 256, 0, stream>>>(
      z_star, NN, dec_W1, HH, dec_b1, amat, nullptr, nullptr, BSZ, HH, NN);
  // 4) r = x - (a @ dec_W2 + b2)
  gemm16<1, false><<<dim3(DD / 32, BSZ / 64), 256, 0, stream>>>(
      amat, HH, dec_W2, DD, dec_b2, rres, x, nullptr, BSZ, DD, HH);
  // 5) c = 2a(1-a^2)(r @ W2^T)/sigma^2
  gemm16<2, true><<<dim3(HH / 32, BSZ / 64), 256, 0, stream>>>(
      rres, DD, dec_W2, DD, nullptr, cvec, amat, sigma, BSZ, HH, DD);
  // 6) Prec per sample
  prec_kernel<<<dim3(BSZ), 256, 0, stream>>>(amat, cvec, dec_W1, dec_W2, sigma,
                                             Prec);
  // 7) eig + cholesky + solve
  eig_chol_kernel<<<dim3(BSZ / 32), 32, 0, stream>>>(Prec, z_star, eps, z_samp,
                                                     lat_e, logd);
  // 8) a2 = tanh(z_sample @ dec_W1 + b1)
  gemm16<0, false><<<dim3(HH / 32, BSZ / 64), 256, 0, stream>>>(
      z_samp, NN, dec_W1, HH, dec_b1, a2, nullptr, nullptr, BSZ, HH, NN);
  // 9) recon-loss partials of (a2 @ W2 + b2 - x)^2
  gemm16<3, false><<<dim3(DD / 32, BSZ / 64), 256, 0, stream>>>(
      a2, HH, dec_W2, DD, dec_b2, partials, x, nullptr, BSZ, DD, HH);
  // 10) assemble outputs
  finalize_kernel<<<dim3(2), 256, 0, stream>>>(partials, sigma, lat_e, logd,
                                               out);
}